// SSMBlock_50362786512963
// MI455X (gfx1250) — compile-verified
//
#include <hip/hip_runtime.h>
#include <hip/hip_bf16.h>

// ---------------------------------------------------------------------------
// Mamba SSM block on gfx1250 (MI455X), fp32 end-to-end.
// GEMMs use V_WMMA_F32_16X16X4_F32 (the only fp32 WMMA shape on CDNA5) to
// keep full precision into the 1024-step selective scan.
// Register blocking: each wave computes a 32x64 (MTILE=2 x NTILE=4) output
// tile so every B fragment is reused across 2 A fragments.
// Addressing is done with running integer offsets off the __restrict__ global
// bases so A/B loads stay on the GLOBAL path (LOADcnt only, no FLAT/DScnt)
// and the inner loop has no 64-bit multiplies.
// ---------------------------------------------------------------------------

typedef __attribute__((ext_vector_type(2))) float v2f;
typedef __attribute__((ext_vector_type(8))) float v8f;

#define BATCH   2
#define LSEQ    1024
#define DMODEL  1024
#define DI      2048      // inner dim = 2*DMODEL
#define NSTATE  16
#define DTR     64
#define DBC_W   96        // DTR + 2*NSTATE

// ---------------------------------------------------------------------------
// 1) LayerNorm: one block per token row of 1024.
// ---------------------------------------------------------------------------
__global__ void ln_kernel(const float* __restrict__ x,
                          const float* __restrict__ g,
                          const float* __restrict__ b,
                          float* __restrict__ h) {
  const int row = blockIdx.x;           // 0..2047
  const int tid = threadIdx.x;          // 0..255
  const float* xr = x + (size_t)row * DMODEL;

  float v[4];
  float sum = 0.f;
#pragma unroll
  for (int i = 0; i < 4; ++i) { v[i] = xr[tid + i * 256]; sum += v[i]; }

  __shared__ float red[256];
  red[tid] = sum; __syncthreads();
  for (int s = 128; s > 0; s >>= 1) {
    if (tid < s) red[tid] += red[tid + s];
    __syncthreads();
  }
  const float mu = red[0] * (1.0f / DMODEL);
  __syncthreads();

  float sq = 0.f;
#pragma unroll
  for (int i = 0; i < 4; ++i) { float d = v[i] - mu; sq += d * d; }
  red[tid] = sq; __syncthreads();
  for (int s = 128; s > 0; s >>= 1) {
    if (tid < s) red[tid] += red[tid + s];
    __syncthreads();
  }
  const float inv = rsqrtf(red[0] * (1.0f / DMODEL) + 1e-5f);

  float* hr = h + (size_t)row * DMODEL;
#pragma unroll
  for (int i = 0; i < 4; ++i) {
    const int c = tid + i * 256;
    hr[c] = (v[i] - mu) * inv * g[c] + b[c];
  }
}

// ---------------------------------------------------------------------------
// 2) Generic fp32 WMMA GEMM: C[M,N] = A[M,K] @ B[K,N] (+ optional residual R).
// One wave computes a (16*MTILE) x (16*NTILE) output tile.
// Fragment layouts per CDNA5 ISA 7.12.2:
//   A (16x4 f32):  lane<16 -> A[row][k..k+1], lane>=16 -> A[row][k+2..k+3]
//   B (4x16 f32):  mirrored, N in lane[3:0]
//   C/D (16x16):   VGPR r -> M = r + 8*(lane>=16), N = lane[3:0]
// ---------------------------------------------------------------------------
template <int MTILE, int NTILE>
__global__ __launch_bounds__(32)
void gemm_f32_wmma(const float* __restrict__ A, int lda,
                   const float* __restrict__ B, int ldb,
                   float* __restrict__ C, int ldc,
                   const float* __restrict__ R, int K) {
  const int lane = threadIdx.x;       // 0..31
  const int half = lane >> 4;         // K-half selector for A/B fragments
  const int lh   = lane & 15;         // M (for A) / N (for B,C,D)
  const int row0 = blockIdx.x * (16 * MTILE);
  const int col0 = blockIdx.y * (16 * NTILE);

  v8f acc[MTILE][NTILE] = {};

  // Running element offsets (keep loads on the GLOBAL path).
  size_t aoff[MTILE];
#pragma unroll
  for (int m = 0; m < MTILE; ++m)
    aoff[m] = (size_t)(row0 + m * 16 + lh) * lda + 2 * half;

  size_t boff = (size_t)(2 * half) * ldb + col0 + lh;
  const size_t bstep = (size_t)4 * ldb;

  for (int k = 0; k < K; k += 4) {
    v2f a[MTILE];
#pragma unroll
    for (int m = 0; m < MTILE; ++m) {
      a[m].x = A[aoff[m]];
      a[m].y = A[aoff[m] + 1];
      aoff[m] += 4;
    }

    v2f bfrag[NTILE];
#pragma unroll
    for (int t = 0; t < NTILE; ++t) {
      bfrag[t].x = B[boff + t * 16];        // B[k + 2*half    ][col]
      bfrag[t].y = B[boff + t * 16 + ldb];  // B[k + 2*half + 1][col]
    }
    boff += bstep;

#pragma unroll
    for (int m = 0; m < MTILE; ++m)
#pragma unroll
      for (int t = 0; t < NTILE; ++t)
        acc[m][t] = __builtin_amdgcn_wmma_f32_16x16x4_f32(
            /*neg_a=*/false, a[m], /*neg_b=*/false, bfrag[t],
            /*c_mod=*/(short)0, acc[m][t],
            /*reuse_a=*/false, /*reuse_b=*/false);
  }

#pragma unroll
  for (int m = 0; m < MTILE; ++m)
#pragma unroll
    for (int t = 0; t < NTILE; ++t)
#pragma unroll
      for (int r = 0; r < 8; ++r) {
        const size_t idx =
            (size_t)(row0 + m * 16 + r + 8 * half) * ldc + col0 + t * 16 + lh;
        float val = acc[m][t][r];
        if (R) val += R[idx];
        C[idx] = val;
      }
}

// ---------------------------------------------------------------------------
// 3) Causal depthwise conv (K=4) + SiLU. x_in = xz[:, :DI].
// ---------------------------------------------------------------------------
__global__ void conv_silu_kernel(const float* __restrict__ xz,
                                 const float* __restrict__ cw,
                                 const float* __restrict__ cb,
                                 float* __restrict__ xc) {
  const int idx = blockIdx.x * blockDim.x + threadIdx.x;  // over B*L*DI = 2^22
  const int di = idx & (DI - 1);
  const int l  = (idx >> 11) & (LSEQ - 1);
  const int b  = idx >> 21;

  float acc = cb[di];
  const float* w = cw + di * 4;
#pragma unroll
  for (int j = 0; j < 4; ++j) {
    const int ls = l - 3 + j;
    if (ls >= 0)
      acc += w[j] * xz[(size_t)(b * LSEQ + ls) * (2 * DI) + di];
  }
  xc[idx] = acc / (1.0f + __expf(-acc));   // silu
}

// ---------------------------------------------------------------------------
// 4) Selective scan. One thread per (batch, channel di); state s[16] in VGPRs.
// Fuses: delta = softplus(delta_pre + b_dt), y = (scan + u*D) * silu(z).
// ---------------------------------------------------------------------------
__global__ void scan_kernel(const float* __restrict__ xc,
                            const float* __restrict__ dpre,
                            const float* __restrict__ dbc,
                            const float* __restrict__ xz,
                            const float* __restrict__ A_log,
                            const float* __restrict__ b_dt,
                            const float* __restrict__ D_param,
                            float* __restrict__ y2) {
  const int di = blockIdx.x * blockDim.x + threadIdx.x;  // 0..2047
  const int b  = blockIdx.y;                             // 0..1

  float Arow[NSTATE], s[NSTATE];
#pragma unroll
  for (int n = 0; n < NSTATE; ++n) {
    Arow[n] = -expf(A_log[di * NSTATE + n]);
    s[n] = 0.f;
  }
  const float bdt = b_dt[di];
  const float Dp  = D_param[di];

  for (int l = 0; l < LSEQ; ++l) {
    const size_t tok = (size_t)(b * LSEQ + l);
    const float u  = xc[tok * DI + di];
    const float dp = dpre[tok * DI + di] + bdt;
    // softplus = max(x,0) + log1p(exp(-|x|))  (stable for all x)
    const float dlt = fmaxf(dp, 0.f) + log1pf(__expf(-fabsf(dp)));

    const float* bc = dbc + tok * DBC_W;  // Bm at [64..79], Cm at [80..95]
    float y = 0.f;
#pragma unroll
    for (int n = 0; n < NSTATE; ++n) {
      const float dA = __expf(dlt * Arow[n]);
      s[n] = dA * s[n] + dlt * bc[DTR + n] * u;
      y += s[n] * bc[DTR + NSTATE + n];
    }

    const float z  = xz[tok * (2 * DI) + DI + di];
    const float sz = z / (1.0f + __expf(-z));
    y2[tok * DI + di] = (y + u * Dp) * sz;
  }
}

// ---------------------------------------------------------------------------
// Launch pipeline.
// ---------------------------------------------------------------------------
extern "C" void kernel_launch(void* const* d_in, const int* in_sizes, int n_in,
                              void* d_out, int out_size, void* d_ws, size_t ws_size,
                              hipStream_t stream) {
  const float* x       = (const float*)d_in[0];
  const float* ln_g    = (const float*)d_in[1];
  const float* ln_b    = (const float*)d_in[2];
  const float* W_in    = (const float*)d_in[3];
  const float* conv_w  = (const float*)d_in[4];
  const float* conv_b  = (const float*)d_in[5];
  const float* W_xproj = (const float*)d_in[6];
  const float* W_dt    = (const float*)d_in[7];
  const float* b_dt    = (const float*)d_in[8];
  const float* A_log   = (const float*)d_in[9];
  const float* D_param = (const float*)d_in[10];
  const float* W_out   = (const float*)d_in[11];
  float* out = (float*)d_out;

  // Workspace carve-up (floats): ~89 MiB total.
  float* ws   = (float*)d_ws;
  float* h    = ws;                                  // [2048,1024]
  float* xz   = h    + (size_t)2048 * 1024;          // [2048,4096]
  float* xc   = xz   + (size_t)2048 * 4096;          // [2048,2048]
  float* dbc  = xc   + (size_t)2048 * 2048;          // [2048,  96]
  float* dpre = dbc  + (size_t)2048 * 96;            // [2048,2048]
  float* y2   = dpre + (size_t)2048 * 2048;          // [2048,2048]

  // 1) LayerNorm
  ln_kernel<<<BATCH * LSEQ, 256, 0, stream>>>(x, ln_g, ln_b, h);

  // 2) xz = h @ W_in        [2048,1024]x[1024,4096]
  gemm_f32_wmma<2, 4><<<dim3(2048 / 32, 4096 / 64), 32, 0, stream>>>(
      h, DMODEL, W_in, 2 * DI, xz, 2 * DI, nullptr, DMODEL);

  // 3) x_c = silu(conv(x_in))
  conv_silu_kernel<<<(BATCH * LSEQ * DI) / 256, 256, 0, stream>>>(
      xz, conv_w, conv_b, xc);

  // 4) dbc = x_c @ W_xproj  [2048,2048]x[2048,96]
  gemm_f32_wmma<2, 2><<<dim3(2048 / 32, 96 / 32), 32, 0, stream>>>(
      xc, DI, W_xproj, DBC_W, dbc, DBC_W, nullptr, DI);

  // 5) delta_pre = dt @ W_dt  (dt = dbc[:, :64], lda=96)
  gemm_f32_wmma<2, 4><<<dim3(2048 / 32, 2048 / 64), 32, 0, stream>>>(
      dbc, DBC_W, W_dt, DI, dpre, DI, nullptr, DTR);

  // 6) selective scan (+softplus, +u*D, *silu(z))
  scan_kernel<<<dim3(DI / 256, BATCH), 256, 0, stream>>>(
      xc, dpre, dbc, xz, A_log, b_dt, D_param, y2);

  // 7) out = x + y2 @ W_out  [2048,2048]x[2048,1024], residual fused
  gemm_f32_wmma<2, 4><<<dim3(2048 / 32, 1024 / 64), 32, 0, stream>>>(
      y2, DI, W_out, DMODEL, out, DMODEL, x, DI);
}